// GATNet_86449101733936
// MI455X (gfx1250) — compile-verified
//
#include <hip/hip_runtime.h>
#include <hip/hip_bf16.h>
#include <math.h>

// ---------------------------------------------------------------------------
// CDNA5 (gfx1250) GAT pipeline.
// Dense GEMMs use V_WMMA_F32_16X16X4_F32 (exact fp32 matrix pipe, wave32),
// register-blocked 16x64 per wave so each A fragment feeds 4 WMMAs.
// All B fragments are loaded into distinct registers BEFORE the WMMA group so
// the loads clause together and the 4 WMMAs issue back-to-back instead of
// load->wait->wmma serialization.
// Edge softmax/aggregation uses f32 global atomics (L2-resident on MI455X:
// whole working set < 192MB L2).
// ---------------------------------------------------------------------------

typedef __attribute__((ext_vector_type(2))) float v2f;
typedef __attribute__((ext_vector_type(8))) float v8f;

#define HID   64
#define NEGS  0.2f
#define BNEPS 1e-5f
#define NGRAPH 128

__device__ __forceinline__ v8f wmma_f32(v2f a, v2f b, v8f c) {
  return __builtin_amdgcn_wmma_f32_16x16x4_f32(
      /*neg_a=*/false, a, /*neg_b=*/false, b,
      /*c_mod=*/(short)0, c, /*reuse_a=*/false, /*reuse_b=*/false);
}

// ---------------------------------------------------------------------------
// fp32 WMMA GEMM:  Z[M,Nout] = A[M,K] @ B[K,Nout]
// Each wave computes a 16x64 output strip (4 accumulators); the A fragment
// is loaded once per K-step and reused across the 4 N-tiles.
// A fragment (16x4 f32, 2 VGPRs): lanes 0-15 hold K=k,k+1; lanes 16-31 K=k+2,k+3
// B fragment (4x16  f32, 2 VGPRs): symmetric (lane%16 = column)
// D 16x16 f32: VGPR r -> row r (lanes 0-15) / row r+8 (lanes 16-31)
// Requires: M % 16 == 0 (50000 = 3125*16), Nout % 64 == 0, K % 4 == 0.
// ---------------------------------------------------------------------------
template <int K>
__global__ void gemm_wmma_f32(const float* __restrict__ A,
                              const float* __restrict__ B,
                              float* __restrict__ Z,
                              int M, int Nout) {
  const int wave   = threadIdx.x >> 5;
  const int lane   = threadIdx.x & 31;
  const int waves  = blockDim.x >> 5;
  const int m0     = blockIdx.x * 16;
  const int n0     = (blockIdx.y * waves + wave) * 64;
  if (m0 >= M || n0 + 64 > Nout) return;   // wave-uniform: EXEC stays all-ones

  const int lane16 = lane & 15;
  const int koff   = (lane >> 4) << 1;     // 0 for lanes 0-15, 2 for 16-31

  const float* arow = A + (long)(m0 + lane16) * K;
  const float* bcol = B + n0 + lane16;

  v8f acc0 = {}, acc1 = {}, acc2 = {}, acc3 = {};
#pragma unroll 4
  for (int k = 0; k < K; k += 4) {
    // ---- load phase: A fragment + all 4 B fragments into distinct regs ----
    v2f a, b0, b1, b2, b3;
    a[0] = arow[k + koff + 0];
    a[1] = arow[k + koff + 1];
    const float* brow0 = bcol + (long)(k + koff + 0) * Nout;
    const float* brow1 = bcol + (long)(k + koff + 1) * Nout;
    b0[0] = brow0[0];   b0[1] = brow1[0];
    b1[0] = brow0[16];  b1[1] = brow1[16];
    b2[0] = brow0[32];  b2[1] = brow1[32];
    b3[0] = brow0[48];  b3[1] = brow1[48];
    // ---- compute phase: 4 back-to-back WMMAs reusing the A fragment ------
    acc0 = wmma_f32(a, b0, acc0);
    acc1 = wmma_f32(a, b1, acc1);
    acc2 = wmma_f32(a, b2, acc2);
    acc3 = wmma_f32(a, b3, acc3);
  }

  const int mbase = m0 + ((lane >> 4) << 3);
  float* zbase = Z + n0 + lane16;
#pragma unroll
  for (int r = 0; r < 8; ++r) {
    float* zp = zbase + (long)(mbase + r) * Nout;
    zp[0]  = acc0[r];
    zp[16] = acc1[r];
    zp[32] = acc2[r];
    zp[48] = acc3[r];
  }
}

// ---------------------------------------------------------------------------
// Per-(node,head) attention coefficients: als = <z, a_src>, ald = <z, a_dst>.
// One wave32 per (node,head); 64 channels -> 2 per lane, shuffle reduce.
// ---------------------------------------------------------------------------
__global__ void attn_coef(const float* __restrict__ z,
                          const float* __restrict__ a_s,
                          const float* __restrict__ a_d,
                          float* __restrict__ als, float* __restrict__ ald,
                          int N, int H) {
  const int wid  = (int)((blockIdx.x * (long)blockDim.x + threadIdx.x) >> 5);
  const int lane = threadIdx.x & 31;
  if (wid >= N * H) return;
  const int n = wid / H, h = wid - n * H;
  const float* zp = z + (long)n * (H * HID) + h * HID;
  const float* sp = a_s + h * HID;
  const float* dp = a_d + h * HID;
  float s = zp[lane] * sp[lane] + zp[lane + 32] * sp[lane + 32];
  float d = zp[lane] * dp[lane] + zp[lane + 32] * dp[lane + 32];
#pragma unroll
  for (int off = 16; off > 0; off >>= 1) {
    s += __shfl_down(s, off, 32);
    d += __shfl_down(d, off, 32);
  }
  if (lane == 0) { als[wid] = s; ald[wid] = d; }
}

__device__ __forceinline__ float atomicMaxFloat(float* addr, float v) {
  if (v >= 0.0f)
    return __int_as_float(atomicMax((int*)addr, __float_as_int(v)));
  else
    return __uint_as_float(atomicMin((unsigned int*)addr, __float_as_uint(v)));
}

__global__ void fill_neg_inf(float* p, int n) {
  int i = blockIdx.x * blockDim.x + threadIdx.x;
  if (i < n) p[i] = -INFINITY;
}

// Pass 1: leaky-relu logits per edge (incl. self-loops), per-dst running max.
__global__ void edge_logit_max(const float* __restrict__ als,
                               const float* __restrict__ ald,
                               const int* __restrict__ src,
                               const int* __restrict__ dst,
                               float* __restrict__ p, float* __restrict__ mmax,
                               int E, int ET, int H) {
  int e = blockIdx.x * blockDim.x + threadIdx.x;
  if (e >= ET) return;
  int s = (e < E) ? src[e] : (e - E);
  int d = (e < E) ? dst[e] : (e - E);
  for (int h = 0; h < H; ++h) {
    float v = als[s * H + h] + ald[d * H + h];
    v = (v > 0.0f) ? v : NEGS * v;
    p[(long)e * H + h] = v;
    atomicMaxFloat(&mmax[d * H + h], v);
  }
}

// Pass 2: p = exp(logit - max[dst]); denom[dst] += p
__global__ void edge_exp_sum(const int* __restrict__ src,
                             const int* __restrict__ dst,
                             const float* __restrict__ mmax,
                             float* __restrict__ p, float* __restrict__ denom,
                             int E, int ET, int H) {
  int e = blockIdx.x * blockDim.x + threadIdx.x;
  if (e >= ET) return;
  int d = (e < E) ? dst[e] : (e - E);
  for (int h = 0; h < H; ++h) {
    float v = __expf(p[(long)e * H + h] - mmax[d * H + h]);
    p[(long)e * H + h] = v;
    atomicAdd(&denom[d * H + h], v);
  }
}

// Pass 3: out[dst] += z[src] * alpha.  One wave per edge, C channels strided.
__global__ void edge_aggregate(const float* __restrict__ z,
                               const float* __restrict__ p,
                               const float* __restrict__ denom,
                               const int* __restrict__ src,
                               const int* __restrict__ dst,
                               float* __restrict__ out,
                               int E, int ET, int H, int C) {
  const int wid  = (int)((blockIdx.x * (long)blockDim.x + threadIdx.x) >> 5);
  const int lane = threadIdx.x & 31;
  if (wid >= ET) return;
  int s = (wid < E) ? src[wid] : (wid - E);
  int d = (wid < E) ? dst[wid] : (wid - E);
  for (int c = lane; c < C; c += 32) {
    int h = c >> 6;  // HID == 64
    float alpha = p[(long)wid * H + h] / (denom[d * H + h] + 1e-16f);
    atomicAdd(&out[(long)d * C + c], z[(long)s * C + c] * alpha);
  }
}

// bias + BatchNorm(eval) + ReLU, in place
__global__ void bias_bn_relu(float* __restrict__ h,
                             const float* __restrict__ b,
                             const float* __restrict__ g,
                             const float* __restrict__ beta,
                             const float* __restrict__ mean,
                             const float* __restrict__ var,
                             long total, int C) {
  long i = blockIdx.x * (long)blockDim.x + threadIdx.x;
  if (i >= total) return;
  int c = (int)(i % C);
  float v = h[i] + b[c];
  v = g[c] * (v - mean[c]) * rsqrtf(var[c] + BNEPS) + beta[c];
  h[i] = (v > 0.0f) ? v : 0.0f;
}

// global mean pool: accumulate sums + counts per graph
__global__ void pool_accum(const float* __restrict__ h,
                           const int* __restrict__ batch,
                           float* __restrict__ pooled, float* __restrict__ cnt,
                           int N, int C) {
  long i = blockIdx.x * (long)blockDim.x + threadIdx.x;
  if (i >= (long)N * C) return;
  int n = (int)(i / C), c = (int)(i - (long)n * C);
  int g = batch[n];
  atomicAdd(&pooled[g * C + c], h[(long)n * C + c]);
  if (c == 0) atomicAdd(&cnt[g], 1.0f);
}

__global__ void fc_kernel(const float* __restrict__ pooled,
                          const float* __restrict__ cnt,
                          const float* __restrict__ fcw,
                          const float* __restrict__ fcb,
                          float* __restrict__ out) {
  int i = blockIdx.x * blockDim.x + threadIdx.x;
  if (i >= NGRAPH * 10) return;
  int g = i / 10, j = i - g * 10;
  float inv = 1.0f / fmaxf(cnt[g], 1.0f);
  float acc = 0.0f;
  for (int c = 0; c < HID; ++c)
    acc += pooled[g * HID + c] * inv * fcw[c * 10 + j];
  out[i] = acc + fcb[j];
}

// ---------------------------------------------------------------------------
extern "C" void kernel_launch(void* const* d_in, const int* in_sizes, int n_in,
                              void* d_out, int out_size, void* d_ws, size_t ws_size,
                              hipStream_t stream) {
  const float* x     = (const float*)d_in[0];
  const int*   ei    = (const int*)d_in[1];
  const int*   batch = (const int*)d_in[2];

  const int F  = 128;
  const int N  = in_sizes[0] / F;       // 50000
  const int E  = in_sizes[1] / 2;       // 800000
  const int ET = E + N;                 // + self loops
  const int* src = ei;
  const int* dst = ei + E;

  const int HEADS[3] = {4, 4, 1};

  // workspace layout (floats)
  float* ws    = (float*)d_ws;
  float* h_buf = ws;                      long o = (long)N * 256;
  float* z_buf = ws + o;                  o += (long)N * 256;
  float* als   = ws + o;                  o += (long)N * 4;
  float* ald   = ws + o;                  o += (long)N * 4;
  float* mmax  = ws + o;                  o += (long)N * 4;
  float* denom = ws + o;                  o += (long)N * 4;
  float* pbuf  = ws + o;                  o += (long)ET * 4;
  float* pooled= ws + o;                  o += (long)NGRAPH * HID;
  float* cnt   = ws + o;                  o += (long)NGRAPH;
  (void)ws_size;

  const float* h_cur = x;
  for (int layer = 0; layer < 3; ++layer) {
    const int H = HEADS[layer];
    const int C = H * HID;
    const float* W    = (const float*)d_in[3 + layer * 8 + 0];
    const float* a_s  = (const float*)d_in[3 + layer * 8 + 1];
    const float* a_d  = (const float*)d_in[3 + layer * 8 + 2];
    const float* bias = (const float*)d_in[3 + layer * 8 + 3];
    const float* bn_g = (const float*)d_in[3 + layer * 8 + 4];
    const float* bn_b = (const float*)d_in[3 + layer * 8 + 5];
    const float* bn_m = (const float*)d_in[3 + layer * 8 + 6];
    const float* bn_v = (const float*)d_in[3 + layer * 8 + 7];

    // z = h @ W  (fp32 WMMA, 16x64 strip per wave)
    {
      int col_waves = C / 64;             // 4 (layers 0,1) or 1 (layer 2)
      dim3 grid((N + 15) / 16, 1);
      dim3 block(32 * col_waves);
      if (layer == 0)
        gemm_wmma_f32<128><<<grid, block, 0, stream>>>(h_cur, W, z_buf, N, C);
      else
        gemm_wmma_f32<256><<<grid, block, 0, stream>>>(h_cur, W, z_buf, N, C);
    }
    // attention coefficients
    {
      int waves = N * H;
      attn_coef<<<(waves * 32 + 255) / 256, 256, 0, stream>>>(
          z_buf, a_s, a_d, als, ald, N, H);
    }
    // init reducers + output accumulator (after GEMM consumed h_buf)
    fill_neg_inf<<<(N * H + 255) / 256, 256, 0, stream>>>(mmax, N * H);
    hipMemsetAsync(denom, 0, (size_t)N * H * sizeof(float), stream);
    hipMemsetAsync(h_buf, 0, (size_t)N * C * sizeof(float), stream);

    // segment softmax over incoming edges
    edge_logit_max<<<(ET + 255) / 256, 256, 0, stream>>>(
        als, ald, src, dst, pbuf, mmax, E, ET, H);
    edge_exp_sum<<<(ET + 255) / 256, 256, 0, stream>>>(
        src, dst, mmax, pbuf, denom, E, ET, H);
    edge_aggregate<<<(int)(((long)ET * 32 + 255) / 256), 256, 0, stream>>>(
        z_buf, pbuf, denom, src, dst, h_buf, E, ET, H, C);

    // bias + BN + ReLU
    {
      long total = (long)N * C;
      bias_bn_relu<<<(int)((total + 255) / 256), 256, 0, stream>>>(
          h_buf, bias, bn_g, bn_b, bn_m, bn_v, total, C);
    }
    h_cur = h_buf;
  }

  // global mean pool + FC
  hipMemsetAsync(pooled, 0, (size_t)NGRAPH * HID * sizeof(float), stream);
  hipMemsetAsync(cnt,    0, (size_t)NGRAPH * sizeof(float), stream);
  pool_accum<<<(int)(((long)N * HID + 255) / 256), 256, 0, stream>>>(
      h_buf, batch, pooled, cnt, N, HID);

  const float* fcw = (const float*)d_in[27];
  const float* fcb = (const float*)d_in[28];
  fc_kernel<<<(NGRAPH * 10 + 255) / 256, 256, 0, stream>>>(
      pooled, cnt, fcw, fcb, (float*)d_out);
}